// SelfAttention_78907139162302
// MI455X (gfx1250) — compile-verified
//
#include <hip/hip_runtime.h>
#include <hip/hip_bf16.h>

// Decode self-attention, MI455X (gfx1250, wave32).
// Bandwidth-bound (~313 MB @ 23.3 TB/s): GEMVs stream weights with b128 loads;
// attention score/context matmuls use V_WMMA_F32_16X16X4_F32 (f32 in/out, no
// precision loss vs the f32 reference). B operands are staged as zero-padded
// 16-row tiles in LDS so WMMA feeds are unconditional ds_load_b64 (no EXEC
// predication); the context-kernel B tile is staged via the CDNA5 async
// global->LDS DMA path (ASYNCcnt).

#define H      4096
#define NH     32
#define NKV    8
#define HD     128
#define T_PREV 16383
#define T_ALL  16384
#define GQA    4        // NH / NKV

typedef __attribute__((ext_vector_type(2))) float v2f;
typedef __attribute__((ext_vector_type(4))) float v4f;
typedef __attribute__((ext_vector_type(8))) float v8f;

__device__ __forceinline__ float waveReduceSum(float v) {
#pragma unroll
  for (int off = 16; off; off >>= 1) v += __shfl_xor(v, off, 32);
  return v;
}

// ---------------------------------------------------------------------------
// Kernel 1: q/k/v projections. One wave per output row; coalesced float4
// streaming of the weight row (the bandwidth driver), shfl reduction.
// rows: [0,4096) -> Wq, [4096,5120) -> Wk, [5120,6144) -> Wv.
// ---------------------------------------------------------------------------
__global__ void qkv_gemv(const float* __restrict__ hidden,
                         const float* __restrict__ Wq,
                         const float* __restrict__ Wk,
                         const float* __restrict__ Wv,
                         float* __restrict__ qraw,
                         float* __restrict__ kraw,
                         float* __restrict__ vraw) {
  const int wave = threadIdx.x >> 5;
  const int lane = threadIdx.x & 31;
  const int row  = blockIdx.x * 8 + wave;   // 0..6143

  const float* W;
  float* out;
  int r;
  if (row < H)                 { W = Wq; out = qraw; r = row; }
  else if (row < H + NKV * HD) { W = Wk; out = kraw; r = row - H; }
  else                         { W = Wv; out = vraw; r = row - H - NKV * HD; }

  const v4f* w4 = reinterpret_cast<const v4f*>(W + (size_t)r * H);
  const v4f* h4 = reinterpret_cast<const v4f*>(hidden);
  float acc = 0.f;
  for (int j = lane; j < H / 4; j += 32) {
    __builtin_prefetch(w4 + j + 256, 0, 1);   // stream-ahead -> global_prefetch
    v4f w = w4[j];
    v4f x = h4[j];
    acc += w.x * x.x + w.y * x.y + w.z * x.z + w.w * x.w;
  }
  acc = waveReduceSum(acc);
  if (lane == 0) out[r] = acc;
}

// ---------------------------------------------------------------------------
// Kernel 2: RoPE on q (2048 pairs) and new k (512 pairs), in place.
// ---------------------------------------------------------------------------
__global__ void rope_kernel(float* __restrict__ q,
                            float* __restrict__ k,
                            const int* __restrict__ pos_ptr) {
  const int QP = H / 2;          // 2048
  const int KP = NKV * HD / 2;   // 512
  int p = blockIdx.x * blockDim.x + threadIdx.x;
  if (p >= QP + KP) return;
  const float pos = (float)pos_ptr[0];

  float* vec;
  int e;
  if (p < QP) { vec = q; e = p * 2; }
  else        { vec = k; e = (p - QP) * 2; }

  const int i = (e & (HD - 1)) >> 1;                    // 0..63
  const float inv_freq = __powf(10000.0f, -(float)i / 64.0f);
  const float f = pos * inv_freq;
  float s, c;
  __sincosf(f, &s, &c);
  const float x0 = vec[e], x1 = vec[e + 1];
  vec[e]     = x0 * c - x1 * s;
  vec[e + 1] = x0 * s + x1 * c;
}

// ---------------------------------------------------------------------------
// Kernel 3: scores[h][t] = (q[h] . k_all[t][h/4]) / sqrt(HD)
// WMMA tile: D(16t x 16n) = A(16t x 4k) * B(4k x 16n), chained over k=0..127.
// A: lane L holds K row t0+(L&15), k pair 2*(L>>4)+{0,1}  (b64 global load).
// B: zero-padded 16x128 q tile in LDS -> unconditional b64 LDS load.
// One wave per 16-key tile; K cache streamed exactly once (64 MB).
// Each active lane owns 8 consecutive t outputs -> two b128 stores.
// ---------------------------------------------------------------------------
__global__ void scores_wmma(const float* __restrict__ kc,    // (T_PREV,NKV,HD)
                            const float* __restrict__ knew,  // (NKV,HD)
                            const float* __restrict__ q,     // (NH,HD) roped
                            float* __restrict__ scores) {    // (NH,T_ALL)
  __shared__ float qs[16 * HD];                   // 8 KB, rows >=4 zeroed
  const int kvh  = blockIdx.x >> 7;               // 128 blocks per kv head
  const int tile = ((blockIdx.x & 127) << 3) + (threadIdx.x >> 5);
  const int lane = threadIdx.x & 31;

  for (int i = threadIdx.x; i < 16 * HD; i += 256) {
    const int n = i >> 7;                         // row 0..15
    qs[i] = (n < GQA) ? q[(kvh * GQA + n) * HD + (i & (HD - 1))] : 0.f;
  }
  __syncthreads();

  const int t0   = tile * 16;
  const int half = lane >> 4;         // k-pair selector
  const int mrow = lane & 15;         // A row (key index in tile)
  const int nn   = lane & 15;         // B/D column (q head in group)

  const int t = t0 + mrow;
  const float* krow = (t < T_PREV) ? (kc + (size_t)t * (NKV * HD) + kvh * HD)
                                   : (knew + kvh * HD);
  v8f acc = {};
#pragma unroll 4
  for (int k0 = 0; k0 < HD; k0 += 4) {
    v2f a = *reinterpret_cast<const v2f*>(krow + k0 + 2 * half);
    v2f b = *reinterpret_cast<const v2f*>(&qs[nn * HD + k0 + 2 * half]);
    acc = __builtin_amdgcn_wmma_f32_16x16x4_f32(false, a, false, b,
                                                (short)0, acc, false, false);
  }

  if (nn < GQA) {
    const float sc = 0.08838834764831845f;        // 1/sqrt(128)
    const int h = kvh * GQA + nn;
    v4f s0 = {acc[0] * sc, acc[1] * sc, acc[2] * sc, acc[3] * sc};
    v4f s1 = {acc[4] * sc, acc[5] * sc, acc[6] * sc, acc[7] * sc};
    float* dst = scores + (size_t)h * T_ALL + t0 + 8 * half;
    *reinterpret_cast<v4f*>(dst)     = s0;        // D: M = r + 8*(lane/16)
    *reinterpret_cast<v4f*>(dst + 4) = s1;
  }
}

// ---------------------------------------------------------------------------
// Kernel 4: per-head softmax. Stores numerators p = exp(s - max) back into
// the scores buffer; per-head sum goes to sums[h] (1/sum folded in later).
// ---------------------------------------------------------------------------
__global__ void softmax_kernel(float* __restrict__ scores,
                               float* __restrict__ sums) {
  __shared__ float red[256];
  const int h   = blockIdx.x;
  const int tid = threadIdx.x;
  float* srow = scores + (size_t)h * T_ALL;

  float m = -3.4e38f;
  for (int t = tid; t < T_ALL; t += 256) m = fmaxf(m, srow[t]);
  red[tid] = m;
  __syncthreads();
  for (int s = 128; s; s >>= 1) {
    if (tid < s) red[tid] = fmaxf(red[tid], red[tid + s]);
    __syncthreads();
  }
  m = red[0];
  __syncthreads();

  float sum = 0.f;
  for (int t = tid; t < T_ALL; t += 256) {
    const float p = __expf(srow[t] - m);
    srow[t] = p;
    sum += p;
  }
  red[tid] = sum;
  __syncthreads();
  for (int s = 128; s; s >>= 1) {
    if (tid < s) red[tid] += red[tid + s];
    __syncthreads();
  }
  if (tid == 0) sums[h] = red[0];
}

// ---------------------------------------------------------------------------
// Kernel 5: ctx[h][d] = sum_t p[h][t] * v_all[t][h/4][d] / sum[h]
// Block = one (kv head, 512-key chunk); its 8 waves split the 8 d-tiles and
// share one zero-padded 16x512 p-tile in LDS, staged with
// global_load_async_to_lds_b128 (ASYNCcnt) overlapped with the zero fill.
// WMMA tile: D(16d x 16n) = A(16d x 4t) * B(4t x 16n), 128 chained WMMAs.
// A loads are coalesced on d; V is streamed exactly once (64 MB).
// ---------------------------------------------------------------------------
__global__ void ctx_wmma(const float* __restrict__ vc,     // (T_PREV,NKV,HD)
                         const float* __restrict__ vnew,   // (NKV,HD)
                         const float* __restrict__ p,      // (NH,T_ALL)
                         const float* __restrict__ sums,   // (NH)
                         float* __restrict__ ctx) {        // (NH,HD) zeroed
  __shared__ float ps[16 * 512];                 // 32 KB, rows >=4 zeroed
  const int kvh   = blockIdx.x >> 5;             // 8 kv heads
  const int chunk = blockIdx.x & 31;             // 32 chunks of 512 keys
  const int wave  = threadIdx.x >> 5;            // d-tile 0..7
  const int lane  = threadIdx.x & 31;
  const int tbase = chunk * 512;
  const int d0    = wave * 16;

  // Async DMA: 4 rows x 512 floats (8 KB) global->LDS, b128 per lane.
  {
    const unsigned psBase = (unsigned)(size_t)ps;         // LDS byte offset
    int g = threadIdx.x;                                  // 2 x 256 x 4 floats
#pragma unroll
    for (int it = 0; it < 2; ++it) {
      const int n   = g >> 7;                             // row 0..3
      const int col = (g & 127) * 4;
      const unsigned ldsoff = psBase + (unsigned)((n * 512 + col) * 4);
      const float* src = p + (size_t)(kvh * GQA + n) * T_ALL + tbase + col;
      asm volatile("global_load_async_to_lds_b128 %0, %1, off"
                   :: "v"(ldsoff), "v"((unsigned long long)(size_t)src)
                   : "memory");
      g += 256;
    }
    for (int i = threadIdx.x; i < 12 * 512; i += 256) ps[4 * 512 + i] = 0.f;
    asm volatile("s_wait_asynccnt 0x0" ::: "memory");
  }
  __syncthreads();

  const int half = lane >> 4;
  const int dl   = lane & 15;          // A row (d)
  const int nn   = lane & 15;          // B/D column (q head)
  const bool act = nn < GQA;
  const int  nq  = nn & 3;

  const float inv = 1.0f / sums[kvh * GQA + nq];

  v8f acc = {};
#pragma unroll 4
  for (int tt = 0; tt < 512; tt += 4) {
    const int tl = tt + 2 * half;                 // lane's two t's: tl, tl+1
    const int ta = tbase + tl;
    const float* v0 = (ta     < T_PREV) ? (vc + (size_t)ta * (NKV * HD) + kvh * HD)
                                        : (vnew + kvh * HD);
    const float* v1 = (ta + 1 < T_PREV) ? (vc + (size_t)(ta + 1) * (NKV * HD) + kvh * HD)
                                        : (vnew + kvh * HD);
    v2f a;
    a.x = v0[d0 + dl];
    a.y = v1[d0 + dl];
    v2f b = *reinterpret_cast<const v2f*>(&ps[nn * 512 + tl]);
    acc = __builtin_amdgcn_wmma_f32_16x16x4_f32(false, a, false, b,
                                                (short)0, acc, false, false);
  }

  if (act) {
    const int h = kvh * GQA + nn;
#pragma unroll
    for (int r = 0; r < 8; ++r) {
      const int d = d0 + r + 8 * half;            // D: M = r + 8*(lane/16)
      atomicAdd(&ctx[(size_t)h * HD + d], acc[r] * inv);
    }
  }
}

// ---------------------------------------------------------------------------
// Kernel 6: out = Wo @ ctx  (same streaming GEMV as kernel 1).
// ---------------------------------------------------------------------------
__global__ void out_gemv(const float* __restrict__ Wo,
                         const float* __restrict__ ctx,
                         float* __restrict__ out) {
  const int wave = threadIdx.x >> 5;
  const int lane = threadIdx.x & 31;
  const int row  = blockIdx.x * 8 + wave;

  const v4f* w4 = reinterpret_cast<const v4f*>(Wo + (size_t)row * H);
  const v4f* x4 = reinterpret_cast<const v4f*>(ctx);
  float acc = 0.f;
  for (int j = lane; j < H / 4; j += 32) {
    __builtin_prefetch(w4 + j + 256, 0, 1);
    v4f w = w4[j];
    v4f x = x4[j];
    acc += w.x * x.x + w.y * x.y + w.z * x.z + w.w * x.w;
  }
  acc = waveReduceSum(acc);
  if (lane == 0) out[row] = acc;
}

// ---------------------------------------------------------------------------
// Workspace layout (floats):
//   [0,4096)        q (roped in place)
//   [4096,5120)     k_new
//   [5120,6144)     v_new
//   [6144,6176)     per-head softmax sums
//   [6208,10304)    ctx (NH*HD)
//   [16384,540672)  scores / softmax numerators (NH*T_ALL, 2 MB)
// ---------------------------------------------------------------------------
extern "C" void kernel_launch(void* const* d_in, const int* in_sizes, int n_in,
                              void* d_out, int out_size, void* d_ws, size_t ws_size,
                              hipStream_t stream) {
  (void)in_sizes; (void)n_in; (void)out_size; (void)ws_size;
  const float* hidden  = (const float*)d_in[0];
  const float* k_cache = (const float*)d_in[1];
  const float* v_cache = (const float*)d_in[2];
  const float* Wq      = (const float*)d_in[3];
  const float* Wk      = (const float*)d_in[4];
  const float* Wv      = (const float*)d_in[5];
  const float* Wo      = (const float*)d_in[6];
  const int*   pos     = (const int*)d_in[7];

  float* ws     = (float*)d_ws;
  float* q      = ws;
  float* kn     = ws + 4096;
  float* vn     = ws + 5120;
  float* sums   = ws + 6144;
  float* ctx    = ws + 6208;
  float* scores = ws + 16384;

  qkv_gemv<<<768, 256, 0, stream>>>(hidden, Wq, Wk, Wv, q, kn, vn);
  rope_kernel<<<10, 256, 0, stream>>>(q, kn, pos);
  scores_wmma<<<1024, 256, 0, stream>>>(k_cache, kn, q, scores);
  softmax_kernel<<<32, 256, 0, stream>>>(scores, sums);
  hipMemsetAsync(ctx, 0, H * sizeof(float), stream);
  ctx_wmma<<<256, 256, 0, stream>>>(v_cache, vn, scores, sums, ctx);
  out_gemv<<<512, 256, 0, stream>>>(Wo, ctx, (float*)d_out);
}